// ItemConv_81741817578251
// MI455X (gfx1250) — compile-verified
//
#include <hip/hip_runtime.h>
#include <hip/hip_bf16.h>

typedef float v2f __attribute__((ext_vector_type(2)));
typedef float v8f __attribute__((ext_vector_type(8)));

constexpr int N_ITEMS = 200000;
constexpr int EMB     = 100;
constexpr int LAYERS  = 3;
constexpr int N_EDGES = 6400000;

// ---------------------------------------------------------------------------
// out[i] = 0.25f * emb[i]   (initialize the layer-mean accumulator with x0/4)
// ---------------------------------------------------------------------------
__global__ void init_accum_kernel(const float* __restrict__ emb,
                                  float* __restrict__ out, int total) {
  int i = blockIdx.x * blockDim.x + threadIdx.x;
  if (i < total) out[i] = 0.25f * emb[i];
}

// ---------------------------------------------------------------------------
// y[m, n] = sum_k x[m, k] * W[n, k]    (torch Linear: y = x @ W.T, no bias)
// One wave per 16x16 output tile, V_WMMA_F32_16X16X4_F32, K stepped by 4.
// A (16x4 f32): lane M = lane%16, half selects K {0,1} vs {2,3}  -> float2/lane
// B (4x16 f32): lane N = lane%16, half selects K {0,1} vs {2,3}  -> float2/lane
// C/D (16x16 f32): VGPR v -> row v + 8*(lane>=16), col lane%16
// ---------------------------------------------------------------------------
__global__ void gemm_wmma_f32_kernel(const float* __restrict__ x,
                                     const float* __restrict__ W,
                                     float* __restrict__ y) {
  const int lane = threadIdx.x & 31;
  const int half = lane >> 4;
  const int l16  = lane & 15;
  const int row0 = blockIdx.x * 16;   // M tile (200000 / 16 = 12500 exact)
  const int n0   = blockIdx.y * 16;   // N tile (7 tiles cover 100, last masked)

  const int  wn     = n0 + l16;
  const bool nvalid = (wn < EMB);
  const float bmask = nvalid ? 1.0f : 0.0f;
  // Clamp the W row for out-of-range columns; zero via bmask (no EXEC branch,
  // WMMA requires EXEC all-1s).
  const float* __restrict__ wrow = W + (size_t)(nvalid ? wn : 0) * EMB;
  const float* __restrict__ xrow = x + (size_t)(row0 + l16) * EMB;

  v8f acc = {};
#pragma unroll 5
  for (int k0 = 0; k0 < EMB; k0 += 4) {
    const int k = k0 + 2 * half;                 // even -> 8B aligned
    float2 a2 = *(const float2*)(xrow + k);
    float2 b2 = *(const float2*)(wrow + k);
    v2f a = { a2.x, a2.y };
    v2f b = { b2.x * bmask, b2.y * bmask };
    acc = __builtin_amdgcn_wmma_f32_16x16x4_f32(false, a, false, b,
                                                (short)0, acc, false, false);
  }

  if (nvalid) {
    const size_t colbase = (size_t)row0 * EMB + wn;
#pragma unroll
    for (int v = 0; v < 8; ++v) {
      y[colbase + (size_t)(v + 8 * half) * EMB] = acc[v];
    }
  }
}

// ---------------------------------------------------------------------------
// SpMM scatter: out[rows[e], :] += vals[e] * y[cols[e], :]
// One wave per edge (grid-stride); lane j covers columns j, j+32, j+64, j+96.
// Both y and out (80 MB each) live in L2, so gathers/atomics hit L2.
// unsafeAtomicAdd forces the native global_atomic_add_f32 (no CAS loop).
// ---------------------------------------------------------------------------
__global__ void spmm_scatter_kernel(const float* __restrict__ y,
                                    const float* __restrict__ vals,
                                    const int* __restrict__ rows,
                                    const int* __restrict__ cols,
                                    float* __restrict__ out, int n_edges) {
  const int lane   = threadIdx.x & 31;
  const int wave   = (blockIdx.x * blockDim.x + threadIdx.x) >> 5;
  const int nwaves = (gridDim.x * blockDim.x) >> 5;

  for (int e = wave; e < n_edges; e += nwaves) {
    const float v = vals[e];
    const float* __restrict__ src = y   + (size_t)cols[e] * EMB;
    float* __restrict__       dst = out + (size_t)rows[e] * EMB;
#pragma unroll
    for (int j = 0; j < 4; ++j) {
      const int col = lane + 32 * j;
      if (col < EMB) {
        unsafeAtomicAdd(dst + col, v * src[col]);
      }
    }
  }
}

// ---------------------------------------------------------------------------
// out[r, :] += 0.25 * x[r, :] / max(||x[r, :]||_2, 1e-12)
// One wave per row; wave-wide shuffle reduction (wave32).
// ---------------------------------------------------------------------------
__global__ void norm_accum_kernel(const float* __restrict__ x,
                                  float* __restrict__ out, int n_rows) {
  const int lane = threadIdx.x & 31;
  const int row  = (blockIdx.x * blockDim.x + threadIdx.x) >> 5;
  if (row >= n_rows) return;

  const float* __restrict__ src = x + (size_t)row * EMB;
  float ss = 0.0f;
#pragma unroll
  for (int j = 0; j < 4; ++j) {
    const int col = lane + 32 * j;
    if (col < EMB) {
      const float t = src[col];
      ss += t * t;
    }
  }
#pragma unroll
  for (int off = 16; off > 0; off >>= 1) ss += __shfl_xor(ss, off, 32);

  const float norm = fmaxf(sqrtf(ss), 1e-12f);
  const float s    = 0.25f / norm;

  float* __restrict__ dst = out + (size_t)row * EMB;
#pragma unroll
  for (int j = 0; j < 4; ++j) {
    const int col = lane + 32 * j;
    if (col < EMB) dst[col] += s * src[col];
  }
}

// ---------------------------------------------------------------------------
// inputs: 0 embedding [N,100] f32, 1 adj_vals [E] f32, 2 weights [3,100,100] f32,
//         3 adj_rows [E] i32, 4 adj_cols [E] i32
// output: [N,100] f32 = mean(x0, n1, n2, n3)
// workspace: 2 x N*EMB floats (160 MB) ping-pong
// ---------------------------------------------------------------------------
extern "C" void kernel_launch(void* const* d_in, const int* in_sizes, int n_in,
                              void* d_out, int out_size, void* d_ws, size_t ws_size,
                              hipStream_t stream) {
  const float* emb   = (const float*)d_in[0];
  const float* vals  = (const float*)d_in[1];
  const float* wts   = (const float*)d_in[2];
  const int*   rows  = (const int*)d_in[3];
  const int*   cols  = (const int*)d_in[4];
  float*       out   = (float*)d_out;

  float* bufA = (float*)d_ws;                       // GEMM output (gather src)
  float* bufB = bufA + (size_t)N_ITEMS * EMB;       // SpMM output (raw x_l)

  const int total = N_ITEMS * EMB;                  // 20,000,000

  // out = 0.25 * x0  (layer-0 copy is NOT normalized)
  init_accum_kernel<<<(total + 255) / 256, 256, 0, stream>>>(emb, out, total);

  const float* x = emb;
  for (int l = 0; l < LAYERS; ++l) {
    // y = x @ W_l.T  via f32 WMMA
    dim3 g(N_ITEMS / 16, (EMB + 15) / 16);          // 12500 x 7 waves
    gemm_wmma_f32_kernel<<<g, 32, 0, stream>>>(x, wts + (size_t)l * EMB * EMB, bufA);

    // zero the scatter target (previous raw x already consumed by the GEMM)
    hipMemsetAsync(bufB, 0, (size_t)total * sizeof(float), stream);

    // x_{l} = segment_sum(vals * y[cols], rows)
    spmm_scatter_kernel<<<4096, 256, 0, stream>>>(bufA, vals, rows, cols, bufB, N_EDGES);

    // out += 0.25 * normalize(x_l);  carried x stays raw
    norm_accum_kernel<<<(N_ITEMS * 32 + 255) / 256, 256, 0, stream>>>(bufB, out, N_ITEMS);

    x = bufB;
  }
}